// mk_16793322127489
// MI455X (gfx1250) — compile-verified
//
#include <hip/hip_runtime.h>

// MI455X / gfx1250, wave32. One wave per batch element, 8 waves/block.
// d_in order (from setup_inputs): c, k, dt, n, nx, detwei
// d_out: nn (B*100 floats) then b (B*10 floats), concatenated flat.
//
// Padding strategy: natural stride-10 row layouts with zero-filled LDS tails.
// Lanes mrow>=10 read finite "wrong-row" values -> pollute only D rows/cols
// >= 10 (never stored). K-padding rows read exact zeros (zero tails + zero
// weights), so the valid D[i<10][j<10] region is exact. b = m*c is a 4-step
// WMMA chain whose B has only column 0 nonzero; zero-padded c kills all junk
// m-columns j>=10.

typedef __attribute__((ext_vector_type(2))) float v2f;
typedef __attribute__((ext_vector_type(4))) float v4f;
typedef __attribute__((ext_vector_type(8))) float v8f;

#define B_TOTAL 200000
#define NGI     13
#define NLOC    10
#define NXROWS  (2 * NGI)          // 26 stacked nx rows
#define WAVES   8                  // waves per block (256 threads)

__global__ __launch_bounds__(256)
void fem_wmma_kernel(const float* __restrict__ c,
                     const float* __restrict__ kp,
                     const float* __restrict__ dtp,
                     const float* __restrict__ n,
                     const float* __restrict__ nx,
                     const float* __restrict__ detwei,
                     float* __restrict__ out_nn,
                     float* __restrict__ out_b)
{
    __shared__ float s_n[176];                 // n, linear, zero tail 130..175 (block)
    __shared__ v4f   s_x4[WAVES][72];          // nx, linear 260 + zero tail to 288
    __shared__ float s_wn[WAVES][16];          // dw,      zeros at 13..15   (nn)
    __shared__ float s_w2[WAVES][32];          // -k*dt*dw duplicated, zeros 26..31
    __shared__ float s_c[WAVES][16];           // c, zero-padded
    __shared__ v4f   s_m4[WAVES][44];          // compact stride-10 tile, 176 floats

    const int tid  = threadIdx.x;
    const int wv   = tid >> 5;
    const int lane = tid & 31;

    int b = blockIdx.x * WAVES + wv;
    if (b >= B_TOTAL) b = B_TOTAL - 1;         // B % 8 == 0: never taken

    const float coef = -kp[0] * dtp[0];        // scalar loads, folded into s_w2

    // ---------------- stage n: linear copy + zero tail ----------------
    if (tid < 176) {
        float v = 0.f;
        if (tid < NGI * NLOC) v = n[tid];
        s_n[tid] = v;
    }

    // ---------- stage nx: coalesced b128 global loads -> linear b128 LDS stores ----------
    {
        const v4f* src4 = (const v4f*)(nx + (size_t)b * (NXROWS * NLOC)); // 16B aligned
        #pragma unroll
        for (int t4 = lane; t4 < 72; t4 += 32) {            // 65 data + 7 zero slots
            v4f v = {0.f, 0.f, 0.f, 0.f};
            if (t4 < (NXROWS * NLOC) / 4) v = src4[t4];
            s_x4[wv][t4] = v;
        }
    }

    // ------------- stage weights, padded c, and zero tail of the m-tile -------------
    {
        int gg = lane;
        if (gg >= NGI) gg -= NGI;                           // 13..25 -> 0..12
        float dwv = (lane < NXROWS) ? detwei[(size_t)b * NGI + gg] : 0.f;
        s_w2[wv][lane] = coef * dwv;                        // lanes 26..31 -> 0
        if (lane < 16) {
            s_wn[wv][lane] = (lane < NGI) ? dwv : 0.f;
            s_c[wv][lane]  = (lane < NLOC) ? c[(size_t)b * NLOC + lane] : 0.f;
        }
        if (lane < 4) {                                     // zero slots 40..43
            v4f z = {0.f, 0.f, 0.f, 0.f};                   // (floats 160..175)
            s_m4[wv][40 + lane] = z;
        }
    }
    __syncthreads();

    const int h    = lane >> 4;        // half-wave select
    const int mrow = lane & 15;        // == A-operand M row == B-operand N col
    const float* s_x  = (const float*)&s_x4[wv][0];
    float*       s_mf = (float*)&s_m4[wv][0];

    // ---------------- nn = N^T diag(w) N : 4 x v_wmma_f32_16x16x4_f32 ----------------
    v8f acc = {0.f, 0.f, 0.f, 0.f, 0.f, 0.f, 0.f, 0.f};
    #pragma unroll
    for (int kk = 0; kk < 4; ++kk) {
        const int ga = 4 * kk + 2 * h;
        const int gb = ga + 1;
        const float na = s_n[ga * NLOC + mrow];
        const float nb = s_n[gb * NLOC + mrow];
        const float wa = s_wn[wv][ga];                 // contiguous pair -> b64
        const float wb = s_wn[wv][gb];
        v2f av, bv;
        av[0] = wa * na;  av[1] = wb * nb;             // A[m,g] = w[g]*n[g,m]
        bv[0] = na;       bv[1] = nb;                  // B[g,j] = n[g,j]
        acc = __builtin_amdgcn_wmma_f32_16x16x4_f32(false, av, false, bv,
                                                    (short)0, acc, false, false);
    }

    // ------- store nn: compact scatter (mrow<10), then linear b128 gather/store -------
    if (mrow < NLOC) {
        #pragma unroll
        for (int v = 0; v < 8; ++v)
            s_mf[(v + 8 * h) * NLOC + mrow] = acc[v];  // rows>=10 land in pad 100..159
    }
    asm volatile("s_wait_dscnt 0" ::: "memory");       // same-wave DS in order; fence
    if (lane < 25) {                                   // 25 x 16B = 400B contiguous
        v4f o = s_m4[wv][lane];
        *(v4f*)(out_nn + (size_t)b * (NLOC * NLOC) + 4 * lane) = o;
    }

    // ------- m = nn - k*dt * X^T diag(w,w) X : 7 more WMMA k-steps, C = nn acc -------
    v8f acc2 = acc;
    #pragma unroll
    for (int kk = 0; kk < 7; ++kk) {
        const int ga = 4 * kk + 2 * h;
        const int gb = ga + 1;
        const float xa = s_x[ga * NLOC + mrow];
        const float xb = s_x[gb * NLOC + mrow];
        const float wa = s_w2[wv][ga];                 // already scaled by -k*dt
        const float wb = s_w2[wv][gb];
        v2f av, bv;
        av[0] = wa * xa;  av[1] = wb * xb;
        bv[0] = xa;       bv[1] = xb;
        acc2 = __builtin_amdgcn_wmma_f32_16x16x4_f32(false, av, false, bv,
                                                     (short)0, acc2, false, false);
    }

    // ---------------- b = m * c : bounce m through LDS, 4 WMMA k-steps ----------------
    if (mrow < NLOC) {
        #pragma unroll
        for (int v = 0; v < 8; ++v)
            s_mf[(v + 8 * h) * NLOC + mrow] = acc2[v]; // overwrite tile (DS in-order)
    }
    asm volatile("s_wait_dscnt 0" ::: "memory");

    const bool is0 = (mrow == 0);                      // only column 0 of B nonzero
    v8f accb = {0.f, 0.f, 0.f, 0.f, 0.f, 0.f, 0.f, 0.f};
    #pragma unroll
    for (int kk = 0; kk < 4; ++kk) {
        const int ga = 4 * kk + 2 * h;                 // K = j index
        const int gb = ga + 1;
        const float ma = s_mf[mrow * NLOC + ga];       // exact for i<10, j<10
        const float mb = s_mf[mrow * NLOC + gb];       // j>=10 junk killed by c=0
        const float ca = s_c[wv][ga];                  // broadcast pair -> b64
        const float cb = s_c[wv][gb];
        v2f av, bv;
        av[0] = ma;              av[1] = mb;           // A[i,j] = m[i,j]
        bv[0] = is0 ? ca : 0.f;  bv[1] = is0 ? cb : 0.f; // B[j,0] = c[j]
        accb = __builtin_amdgcn_wmma_f32_16x16x4_f32(false, av, false, bv,
                                                     (short)0, accb, false, false);
    }

    // D column 0: lane 0 holds b[0..7], lane 16 holds b[8..9]; b*40 is 8B aligned
    {
        float* ob = out_b + (size_t)b * NLOC;
        if (lane == 0) {
            v2f p;
            p[0] = accb[0]; p[1] = accb[1]; *(v2f*)(ob + 0) = p;
            p[0] = accb[2]; p[1] = accb[3]; *(v2f*)(ob + 2) = p;
            p[0] = accb[4]; p[1] = accb[5]; *(v2f*)(ob + 4) = p;
            p[0] = accb[6]; p[1] = accb[7]; *(v2f*)(ob + 6) = p;
        }
        if (lane == 16) {
            v2f p;
            p[0] = accb[0]; p[1] = accb[1]; *(v2f*)(ob + 8) = p;
        }
    }
}

extern "C" void kernel_launch(void* const* d_in, const int* in_sizes, int n_in,
                              void* d_out, int out_size, void* d_ws, size_t ws_size,
                              hipStream_t stream) {
    (void)in_sizes; (void)n_in; (void)out_size; (void)d_ws; (void)ws_size;
    const float* c   = (const float*)d_in[0];
    const float* k   = (const float*)d_in[1];
    const float* dt  = (const float*)d_in[2];
    const float* n   = (const float*)d_in[3];
    const float* nx  = (const float*)d_in[4];
    const float* dw  = (const float*)d_in[5];

    float* out_nn = (float*)d_out;                              // B*100
    float* out_b  = out_nn + (size_t)B_TOTAL * NLOC * NLOC;     // B*10

    const int blocks = (B_TOTAL + WAVES - 1) / WAVES;           // 25000
    fem_wmma_kernel<<<blocks, WAVES * 32, 0, stream>>>(c, k, dt, n, nx, dw,
                                                       out_nn, out_b);
}